// DifferentiableHMM_1924145348739
// MI455X (gfx1250) — compile-verified
//
#include <hip/hip_runtime.h>
#include <math.h>
#include <stdint.h>

typedef __attribute__((ext_vector_type(2))) float v2f;
typedef __attribute__((ext_vector_type(8))) float v8f;
typedef __attribute__((ext_vector_type(4))) unsigned int u32x4;
typedef __attribute__((ext_vector_type(4))) int i32x4;
typedef __attribute__((ext_vector_type(8))) int i32x8;

// Problem dims (fixed by the reference)
constexpr int kNB = 128;          // batch
constexpr int kNS = 512;          // sequence
constexpr int kNK = 64;           // states
constexpr int kNF = 32;           // features
constexpr int kKA = 2 * kNF;      // augmented feature dim [x, x*x] = 64
constexpr int kBS = kNB * kNS;    // 65536 rows for the emission GEMM

// Workspace layout (float offsets)
constexpr size_t P_OFF     = 0;                       // [kKA*kNK]  = 4096
constexpr size_t CONST_OFF = P_OFF + kKA * kNK;       // [kNK]      = 64
constexpr size_t LT_OFF    = CONST_OFF + kNK;         // [kNK*kNK]  = 4096
constexpr size_t LS_OFF    = LT_OFF + kNK * kNK;      // [kNK]      = 64
constexpr size_t ALPHA_OFF = LS_OFF + kNK;            // [kNB*kNK]  = 8192
constexpr size_t EM_OFF    = ALPHA_OFF + kNB * kNK;   // [kBS*kNK]  = 4194304
// total: (EM_OFF + kBS*kNK) * 4 bytes ~= 16.1 MB of d_ws

// ---------------------------------------------------------------------------
// Prep: log_softmax(trans, axis=1), log_softmax(start), emission GEMM params.
//   log_em[r,k] = const_k + sum_f x[r,f]*(mu*w)[k,f] + sum_f x^2[r,f]*(-0.5w)[k,f]
// P is stored [kKA rows (feature) x kNK cols (state)], row-major.
// ---------------------------------------------------------------------------
__global__ __launch_bounds__(64)
void hmm_prep(const float* __restrict__ start_logits,
              const float* __restrict__ trans,
              const float* __restrict__ mu,
              const float* __restrict__ logvar,
              float* __restrict__ ws) {
  const int k = threadIdx.x;  // one thread per state, 0..63
  float* P   = ws + P_OFF;
  float* cst = ws + CONST_OFF;
  float* LT  = ws + LT_OFF;
  float* LS  = ws + LS_OFF;

  // Row-wise log_softmax of trans (axis=1)
  float m = -INFINITY;
  for (int j = 0; j < kNK; ++j) m = fmaxf(m, trans[k * kNK + j]);
  float s = 0.f;
  for (int j = 0; j < kNK; ++j) s += __expf(trans[k * kNK + j] - m);
  const float lse = m + __logf(s);
  for (int j = 0; j < kNK; ++j) LT[k * kNK + j] = trans[k * kNK + j] - lse;

  // log_softmax of start (redundant reduction per thread; K=64 is tiny)
  float ms = -INFINITY;
  for (int j = 0; j < kNK; ++j) ms = fmaxf(ms, start_logits[j]);
  float ss = 0.f;
  for (int j = 0; j < kNK; ++j) ss += __expf(start_logits[j] - ms);
  LS[k] = start_logits[k] - (ms + __logf(ss));

  // Emission parameters
  const float LOG_2PI = 1.83787706640934548356f;
  float c = 0.f;
  for (int f = 0; f < kNF; ++f) {
    const float lv  = logvar[k * kNF + f];
    const float w   = __expf(-lv);
    const float mkf = mu[k * kNF + f];
    P[f * kNK + k]          = mkf * w;     // coefficient of x
    P[(kNF + f) * kNK + k]  = -0.5f * w;   // coefficient of x^2
    c += -0.5f * (LOG_2PI + lv + mkf * mkf * w);
  }
  cst[k] = c;
}

// ---------------------------------------------------------------------------
// Emission GEMM via V_WMMA_F32_16X16X4_F32, with the 16 KB parameter matrix
// staged into LDS by the Tensor Data Mover (one tensor_load_to_lds, 2D D#:
// 64x64 tile of 4-byte elements, stride 64), tracked with TENSORcnt.
// Block = 256 threads = 8 waves; each wave computes a 16(row) x 64(state)
// tile of log_em: 16 K-chunks of 4 x 4 N-tiles = 64 WMMAs per wave.
// ---------------------------------------------------------------------------
__global__ __launch_bounds__(256)
void hmm_emission_wmma(const float* __restrict__ x,
                       const float* __restrict__ ws,
                       float* __restrict__ em) {
  __shared__ float sP[kKA * kNK];   // 16 KB parameter matrix
  __shared__ float sX[8][16 * kNF]; // per-wave 16x32 x-tile, 16 KB total

  const int tid  = threadIdx.x;
  const int wave = tid >> 5;
  const int lane = tid & 31;
  const int half = lane >> 4;   // 0 | 1: selects K sub-pair and M half
  const int l16  = lane & 15;
  const int r0   = blockIdx.x * 128 + wave * 16;  // first row of this wave's tile

  // --- TDM: DMA the whole P matrix (64x64 f32 = 16 KB) into LDS -----------
  if (wave == 0) {
    const unsigned long long ga =
        (unsigned long long)(uintptr_t)(ws + P_OFF);          // global tile addr
    const unsigned lds_addr =
        (unsigned)(uintptr_t)(&sP[0]);                        // LDS byte addr

    u32x4 g0;
    g0[0] = 1u;                                   // count=1, user mode, no gather
    g0[1] = lds_addr;                             // lds_addr [63:32]
    g0[2] = (unsigned)(ga & 0xffffffffull);       // global_addr[31:0]
    g0[3] = (unsigned)((ga >> 32) & 0x01ffffffull)// global_addr[56:32]
            | 0x80000000u;                        // type=2 ("image") [127:126]

    i32x8 g1;
    g1[0] = 0x00020000;   // workgroup_mask=0, data_size=2 (4 bytes)
    g1[1] = 64 << 16;     // tensor_dim0 = 64 (bits 79:48)
    g1[2] = 64 << 16;     // tensor_dim1 = 64 (bits 111:80)
    g1[3] = 64 << 16;     // tile_dim0   = 64 (bits 127:112)
    g1[4] = 64;           // tile_dim1   = 64 (bits 143:128), tile_dim2=0
    g1[5] = 64;           // tensor_dim0_stride = 64 (bits 207:160)
    g1[6] = 0;
    g1[7] = 0;

    i32x4 g2 = {0, 0, 0, 0};
    i32x4 g3 = {0, 0, 0, 0};

#if __clang_major__ >= 23
    i32x8 g4 = {0, 0, 0, 0, 0, 0, 0, 0};
    __builtin_amdgcn_tensor_load_to_lds(g0, g1, g2, g3, g4, 0);
#else
    __builtin_amdgcn_tensor_load_to_lds(g0, g1, g2, g3, 0);
#endif
  }

  // Stage this wave's 16x32 x-tile (coalesced: lanes sweep columns)
  {
    const float* xg = x + (size_t)r0 * kNF;
    float* sx = &sX[wave][0];
#pragma unroll
    for (int i = 0; i < 16; ++i) sx[lane + 32 * i] = xg[lane + 32 * i];
  }
  __builtin_amdgcn_s_wait_tensorcnt(0);   // wave0's DMA complete (no-op elsewhere)
  __syncthreads();

  // Accumulators initialized with per-state constant (depends on N only)
  const float* cst = ws + CONST_OFF;
  v8f acc[4];
#pragma unroll
  for (int n = 0; n < 4; ++n) {
    const float c0 = cst[n * 16 + l16];
#pragma unroll
    for (int r = 0; r < 8; ++r) acc[n][r] = c0;
  }

  const float* sx = &sX[wave][0];
#pragma unroll
  for (int c = 0; c < 16; ++c) {          // K-chunks of 4 over augmented dim
    const int kb = 4 * c + 2 * half;      // this lane-half's first K index
    // A fragment (ISA 7.12.2): lanes 0-15 hold K={kb,kb+1}, row = l16
    v2f a;
    {
      const int k0 = kb, k1 = kb + 1;
      float v0 = (k0 < kNF) ? sx[l16 * kNF + k0] : sx[l16 * kNF + (k0 - kNF)];
      if (k0 >= kNF) v0 = v0 * v0;
      float v1 = (k1 < kNF) ? sx[l16 * kNF + k1] : sx[l16 * kNF + (k1 - kNF)];
      if (k1 >= kNF) v1 = v1 * v1;
      a.x = v0;
      a.y = v1;
    }
#pragma unroll
    for (int n = 0; n < 4; ++n) {
      v2f bfrag;
      bfrag.x = sP[kb * kNK + n * 16 + l16];
      bfrag.y = sP[(kb + 1) * kNK + n * 16 + l16];
      acc[n] = __builtin_amdgcn_wmma_f32_16x16x4_f32(
          /*neg_a=*/false, a, /*neg_b=*/false, bfrag,
          /*c_mod=*/(short)0, acc[n], /*reuse_a=*/false, /*reuse_b=*/false);
    }
  }

  // Store D tile (C/D layout: VGPR r -> rows r and r+8)
#pragma unroll
  for (int n = 0; n < 4; ++n) {
    const int col = n * 16 + l16;
#pragma unroll
    for (int r = 0; r < 8; ++r) {
      const int row = r + 8 * half;
      em[(size_t)(r0 + row) * kNK + col] = acc[n][r];
    }
  }
}

// ---------------------------------------------------------------------------
// Forward recurrence: one batch per block; 128 threads = state j x 2-way
// split of the i-reduction. Each half computes a partial (max, sum-exp) over
// 32 predecessor states; halves are merged with an exact logsumexp combine.
// T, alpha and the partials live in LDS.
// ---------------------------------------------------------------------------
__global__ __launch_bounds__(128)
void hmm_forward(const float* __restrict__ ws,
                 float* __restrict__ alpha_out,
                 const float* __restrict__ em) {
  __shared__ float sT[kNK * kNK];  // 16 KB log-transition matrix
  __shared__ float sA[kNK];        // alpha
  __shared__ float pm[2][kNK];     // partial max
  __shared__ float ps[2][kNK];     // partial shifted sum

  const int tid  = threadIdx.x;
  const int j    = tid & (kNK - 1);   // state column
  const int half = tid >> 6;          // which half of the i-range
  const int b    = blockIdx.x;        // batch
  const float* LT  = ws + LT_OFF;
  const float* LS  = ws + LS_OFF;
  const float* emb = em + (size_t)b * kNS * kNK;

  for (int i = tid; i < kNK * kNK; i += 128) sT[i] = LT[i];
  if (half == 0) sA[j] = LS[j] + emb[j];   // alpha0
  __syncthreads();

  const int i0 = half * (kNK / 2);
  for (int t = 1; t < kNS; ++t) {
    if (t + 4 < kNS)  // software prefetch of upcoming emission rows
      __builtin_prefetch(&emb[(size_t)(t + 4) * kNK + j], 0, 0);

    float m = -INFINITY;
#pragma unroll
    for (int i = 0; i < kNK / 2; ++i)
      m = fmaxf(m, sA[i0 + i] + sT[(i0 + i) * kNK + j]);
    float s = 0.f;
#pragma unroll
    for (int i = 0; i < kNK / 2; ++i)
      s += __expf(sA[i0 + i] + sT[(i0 + i) * kNK + j] - m);
    pm[half][j] = m;
    ps[half][j] = s;
    __syncthreads();                 // partials ready AND old alpha fully read
    if (half == 0) {
      const float m0 = pm[0][j], m1 = pm[1][j];
      const float M  = fmaxf(m0, m1);
      const float S  = ps[0][j] * __expf(m0 - M) + ps[1][j] * __expf(m1 - M);
      sA[j] = M + __logf(S) + emb[(size_t)t * kNK + j];
    }
    __syncthreads();                 // new alpha visible to both halves
  }
  if (half == 0) alpha_out[(size_t)b * kNK + j] = sA[j];
}

// ---------------------------------------------------------------------------
// Final: mean over batches of logsumexp over states -> scalar.
// ---------------------------------------------------------------------------
__global__ __launch_bounds__(128)
void hmm_final(const float* __restrict__ alpha, float* __restrict__ out) {
  __shared__ float red[kNB];
  const int b = threadIdx.x;
  const float* a = alpha + (size_t)b * kNK;
  float m = -INFINITY;
  for (int k = 0; k < kNK; ++k) m = fmaxf(m, a[k]);
  float s = 0.f;
  for (int k = 0; k < kNK; ++k) s += __expf(a[k] - m);
  red[b] = m + __logf(s);
  __syncthreads();
  for (int off = kNB / 2; off > 0; off >>= 1) {
    if (b < off) red[b] += red[b + off];
    __syncthreads();
  }
  if (b == 0) out[0] = red[0] / (float)kNB;
}

// ---------------------------------------------------------------------------
extern "C" void kernel_launch(void* const* d_in, const int* in_sizes, int n_in,
                              void* d_out, int out_size, void* d_ws, size_t ws_size,
                              hipStream_t stream) {
  const float* x            = (const float*)d_in[0];  // [128,512,32]
  const float* start_logits = (const float*)d_in[1];  // [64]
  const float* trans        = (const float*)d_in[2];  // [64,64]
  const float* mu           = (const float*)d_in[3];  // [64,32]
  const float* logvar       = (const float*)d_in[4];  // [64,32]
  float* ws  = (float*)d_ws;                          // needs ~16.1 MB
  float* out = (float*)d_out;
  float* em  = ws + EM_OFF;

  hipLaunchKernelGGL(hmm_prep, dim3(1), dim3(64), 0, stream,
                     start_logits, trans, mu, logvar, ws);
  hipLaunchKernelGGL(hmm_emission_wmma, dim3(kBS / 128), dim3(256), 0, stream,
                     x, ws, em);
  hipLaunchKernelGGL(hmm_forward, dim3(kNB), dim3(128), 0, stream,
                     ws, ws + ALPHA_OFF, em);
  hipLaunchKernelGGL(hmm_final, dim3(1), dim3(128), 0, stream,
                     ws + ALPHA_OFF, out);
}